// K_means_63780264346312
// MI455X (gfx1250) — compile-verified
//
#include <hip/hip_runtime.h>

// K-means (Lloyd) for N=1e6, D=64, K=128, threshold 1.0 on squared mean shift.
// Assignment GEMM runs on V_WMMA_F32_16X16X4_F32 (f32 matrix pipe, full precision).

#define N_PTS   1000000
#define DIM     64
#define KCLUS   128
#define THRESH  1.0f
#define MAX_ITERS 64
#define ASSIGN_BLOCKS 1024
#define MPITCH  68   // padded LDS row pitch for means (bank-conflict-free)

typedef float v2f __attribute__((ext_vector_type(2)));
typedef float v8f __attribute__((ext_vector_type(8)));

// ---------------- init: copy means into workspace, clear flags ----------------
__global__ void k_init(const float* __restrict__ means_in, float* __restrict__ means_ws,
                       int* __restrict__ flags) {
    int i = blockIdx.x * blockDim.x + threadIdx.x;
    int stride = gridDim.x * blockDim.x;
    for (int j = i; j < KCLUS * DIM; j += stride) means_ws[j] = means_in[j];
    if (i == 0) { flags[0] = 0; flags[1] = 0; }   // [0]=converged, [1]=any_big
}

// ---------------- per-iter: zero accumulators ----------------
__global__ void k_begin(float* __restrict__ sums, float* __restrict__ counts,
                        int* __restrict__ flags) {
    if (flags[0]) return;
    int i = blockIdx.x * blockDim.x + threadIdx.x;
    int stride = gridDim.x * blockDim.x;
    for (int j = i; j < KCLUS * DIM; j += stride) sums[j] = 0.0f;
    for (int j = i; j < KCLUS; j += stride) counts[j] = 0.0f;
    if (i == 0) flags[1] = 0;
}

// ---------------- assignment + accumulation ----------------
// Block: 256 threads = 8 waves; each wave owns 16 points; block tile = 128 points.
__launch_bounds__(256)
__global__ void k_assign(const float* __restrict__ Data, const float* __restrict__ means,
                         float* __restrict__ sums_g, float* __restrict__ counts_g,
                         const int* __restrict__ flags) {
    if (flags[0]) return;

    __shared__ float s_means[KCLUS * MPITCH];   // 34816 B
    __shared__ float s_mnorm[KCLUS];
    __shared__ float s_sums[KCLUS * DIM];       // 32768 B
    __shared__ float s_counts[KCLUS];

    const int tid = threadIdx.x;

    // stage means (row-major, padded pitch) + zero block accumulators
    for (int i = tid; i < KCLUS * DIM; i += 256)
        s_means[(i >> 6) * MPITCH + (i & 63)] = means[i];
    for (int i = tid; i < KCLUS * DIM; i += 256) s_sums[i] = 0.0f;
    if (tid < KCLUS) s_counts[tid] = 0.0f;
    __syncthreads();

    if (tid < KCLUS) {
        float s = 0.0f;
        #pragma unroll
        for (int c = 0; c < DIM; ++c) { float m = s_means[tid * MPITCH + c]; s += m * m; }
        s_mnorm[tid] = s;
    }
    __syncthreads();

    const int lane    = tid & 31;
    const int wv      = tid >> 5;
    const int halfsel = lane >> 4;        // 0: K-chunk lanes hold K+0,K+1 ; 1: K+2,K+3
    const int lrow    = lane & 15;        // M (A) / N (B) index within tile
    const int kofs    = halfsel * 2;

    const int ntiles = (N_PTS + 127) / 128;
    for (int tile = blockIdx.x; tile < ntiles; tile += gridDim.x) {
        const int prow = tile * 128 + wv * 16 + lrow;           // this lane's point
        const int crow = prow < N_PTS ? prow : N_PTS - 1;       // clamp for safe loads

        // A fragments: 16 K-chunks of 4, this lane holds dims {4ck+kofs, 4ck+kofs+1}
        const float* aptr = Data + (size_t)crow * DIM + kofs;
        v2f afrag[16];
        #pragma unroll
        for (int ck = 0; ck < 16; ++ck)
            afrag[ck] = *(const v2f*)(aptr + 4 * ck);

        float bestv[8];
        int   besti[8];
        #pragma unroll
        for (int r = 0; r < 8; ++r) { bestv[r] = 3.4e38f; besti[r] = 0; }

        // 8 cluster tiles of 16; dist = ||m||^2 - 2 * (x . m)
        #pragma unroll
        for (int t = 0; t < 8; ++t) {
            v8f acc = {0.f, 0.f, 0.f, 0.f, 0.f, 0.f, 0.f, 0.f};
            const float* bptr = &s_means[(t * 16 + lrow) * MPITCH + kofs];
            #pragma unroll
            for (int ck = 0; ck < 16; ++ck) {
                v2f bfrag = *(const v2f*)(bptr + 4 * ck);
                acc = __builtin_amdgcn_wmma_f32_16x16x4_f32(
                        false, afrag[ck], false, bfrag, (short)0, acc, false, false);
            }
            const float mn = s_mnorm[t * 16 + lrow];
            const int   j  = t * 16 + lrow;
            #pragma unroll
            for (int r = 0; r < 8; ++r) {
                float dist = fmaf(-2.0f, acc[r], mn);
                if (dist < bestv[r]) { bestv[r] = dist; besti[r] = j; }
            }
        }

        // per-point argmin across the 16 lanes of each half-wave
        int mylab = 0;
        #pragma unroll
        for (int r = 0; r < 8; ++r) {
            float v  = bestv[r];
            int   id = besti[r];
            #pragma unroll
            for (int m = 1; m < 16; m <<= 1) {
                float ov = __shfl_xor(v, m, 32);
                int   oi = __shfl_xor(id, m, 32);
                if (ov < v || (ov == v && oi < id)) { v = ov; id = oi; }
            }
            // lanes 0-15 now hold point r's label; lanes 16-31 hold point r+8's
            int xid      = __shfl_xor(id, 16, 32);
            int lab_low  = halfsel ? xid : id;   // label of point r
            int lab_high = halfsel ? id  : xid;  // label of point r+8
            if (lrow == r)     mylab = lab_low;
            if (lrow == r + 8) mylab = lab_high;
        }

        // accumulate into block-local sums (each lane owns 32 dims of its point)
        if (prow < N_PTS) {
            if (lane < 16) atomicAdd(&s_counts[mylab], 1.0f);
            float* dst = &s_sums[mylab * DIM + kofs];
            #pragma unroll
            for (int ck = 0; ck < 16; ++ck) {
                atomicAdd(&dst[4 * ck],     afrag[ck].x);
                atomicAdd(&dst[4 * ck + 1], afrag[ck].y);
            }
        }
    }

    __syncthreads();
    for (int i = tid; i < KCLUS * DIM; i += 256) atomicAdd(&sums_g[i], s_sums[i]);
    if (tid < KCLUS) atomicAdd(&counts_g[tid], s_counts[tid]);
}

// ---------------- mean update + convergence delta ----------------
__global__ void k_update(float* __restrict__ means, const float* __restrict__ sums,
                         const float* __restrict__ counts, int* __restrict__ flags) {
    if (flags[0]) return;
    const int j = blockIdx.x;        // cluster
    const int d = threadIdx.x;       // dim (64 threads = 2 waves)
    float cnt  = counts[j];
    float oldv = means[j * DIM + d];
    float newv = (cnt > 0.0f) ? sums[j * DIM + d] / fmaxf(cnt, 1.0f) : oldv;
    float diff = oldv - newv;
    float sq   = diff * diff;
    #pragma unroll
    for (int m = 1; m < 32; m <<= 1) sq += __shfl_xor(sq, m, 32);
    __shared__ float part[2];
    if ((threadIdx.x & 31) == 0) part[threadIdx.x >> 5] = sq;
    __syncthreads();
    if (threadIdx.x == 0) {
        float delta = part[0] + part[1];
        if (delta > THRESH) atomicOr(&flags[1], 1);
    }
    means[j * DIM + d] = newv;
}

// ---------------- latch convergence ----------------
__global__ void k_end(int* __restrict__ flags) {
    if (threadIdx.x == 0) {
        if (!flags[0] && flags[1] == 0) flags[0] = 1;
    }
}

// ---------------- emit result ----------------
__global__ void k_out(const float* __restrict__ means, float* __restrict__ out) {
    int i = blockIdx.x * blockDim.x + threadIdx.x;
    if (i < KCLUS * DIM) out[i] = means[i];
}

extern "C" void kernel_launch(void* const* d_in, const int* in_sizes, int n_in,
                              void* d_out, int out_size, void* d_ws, size_t ws_size,
                              hipStream_t stream) {
    const float* Data     = (const float*)d_in[0];
    const float* means_in = (const float*)d_in[1];

    float* ws       = (float*)d_ws;
    float* means_ws = ws;                       // 8192 floats
    float* sums     = ws + KCLUS * DIM;         // 8192 floats
    float* counts   = ws + 2 * KCLUS * DIM;     // 128 floats
    int*   flags    = (int*)(ws + 2 * KCLUS * DIM + KCLUS);  // 2 ints

    k_init<<<32, 256, 0, stream>>>(means_in, means_ws, flags);

    for (int it = 0; it < MAX_ITERS; ++it) {
        k_begin<<<32, 256, 0, stream>>>(sums, counts, flags);
        k_assign<<<ASSIGN_BLOCKS, 256, 0, stream>>>(Data, means_ws, sums, counts, flags);
        k_update<<<KCLUS, 64, 0, stream>>>(means_ws, sums, counts, flags);
        k_end<<<1, 32, 0, stream>>>(flags);
    }

    k_out<<<32, 256, 0, stream>>>(means_ws, (float*)d_out);
}